// Affinity_61615600829060
// MI455X (gfx1250) — compile-verified
//
#include <hip/hip_runtime.h>
#include <hip/hip_bf16.h>

// Problem constants (from reference): BATCH=16, D=128, N=16, E=120
#define BATCHN 16
#define DD 128
#define NN 16
#define EE 120

typedef float v2f __attribute__((ext_vector_type(2)));
typedef float v8f __attribute__((ext_vector_type(8)));

__device__ __forceinline__ v8f wmma4(v2f a, v2f b, v8f c) {
  // V_WMMA_F32_16X16X4_F32 : D(16x16 f32) = A(16x4 f32) * B(4x16 f32) + C
  return __builtin_amdgcn_wmma_f32_16x16x4_f32(
      /*neg_a=*/false, a, /*neg_b=*/false, b,
      /*c_mod=*/(short)0, c, /*reuse_a=*/false, /*reuse_b=*/false);
}

// ---------------------------------------------------------------------------
// Kernel 1: l1 = relu(lam1 + lam1^T), l2 = relu(lam2 + lam2^T)   (128x128 each)
// ---------------------------------------------------------------------------
__global__ void __launch_bounds__(256)
prep_weights_kernel(const float* __restrict__ lam1, const float* __restrict__ lam2,
                    float* __restrict__ l1, float* __restrict__ l2) {
  int idx = blockIdx.x * 256 + threadIdx.x;   // 0..16383
  int r = idx >> 7, c = idx & 127;
  float v1 = lam1[r * DD + c] + lam1[c * DD + r];
  float v2 = lam2[r * DD + c] + lam2[c * DD + r];
  l1[idx] = fmaxf(v1, 0.0f);
  l2[idx] = fmaxf(v2, 0.0f);
}

// ---------------------------------------------------------------------------
// Kernel 2: per batch (one workgroup = 8 wave32), compute with f32 WMMA:
//   T1 = l1 @ F_tgt[b], T2 = l2 @ F_tgt[b]          (128x16, in LDS)
//   P1 = F_src[b]^T @ T1, P2 = F_src[b]^T @ T2       (16x16 -> tables)
//   Mp = U_src[b]^T @ U_tgt[b]                       (16x16 -> tables)
//   EM[p][q] = (p<q) && A_src[e]>0 && A_tgt[e]>0     (16x16 -> tables)
// tables layout per batch (1024 floats): [P1(256) | P2(256) | Mp(256) | EM(256)]
// ---------------------------------------------------------------------------
__global__ void __launch_bounds__(256)
batch_tables_kernel(const int* __restrict__ A_src, const int* __restrict__ A_tgt,
                    const float* __restrict__ F_src, const float* __restrict__ F_tgt,
                    const float* __restrict__ U_src, const float* __restrict__ U_tgt,
                    const float* __restrict__ l1g, const float* __restrict__ l2g,
                    float* __restrict__ tables) {
  __shared__ __align__(16) float sFs[DD * NN];
  __shared__ __align__(16) float sFt[DD * NN];
  __shared__ __align__(16) float sT1[DD * NN];
  __shared__ __align__(16) float sT2[DD * NN];
  __shared__ __align__(16) float sUs[NN * NN];
  __shared__ __align__(16) float sUt[NN * NN];

  const int b    = blockIdx.x;
  const int tid  = threadIdx.x;
  const int lane = tid & 31;
  const int wave = tid >> 5;
  const int lrow = lane & 15;   // M (A) / N (B,C/D) position
  const int hi   = lane >> 4;   // half-wave select (K pair / row+8)

  // Stage F matrices into LDS (row-major [d][n], contiguous in global)
  const float4* fs4 = reinterpret_cast<const float4*>(F_src + b * (DD * NN));
  const float4* ft4 = reinterpret_cast<const float4*>(F_tgt + b * (DD * NN));
  float4* sFs4 = reinterpret_cast<float4*>(sFs);
  float4* sFt4 = reinterpret_cast<float4*>(sFt);
  for (int idx = tid; idx < (DD * NN) / 4; idx += 256) {
    sFs4[idx] = fs4[idx];
    sFt4[idx] = ft4[idx];
  }
  sUs[tid] = U_src[b * (NN * NN) + tid];
  sUt[tid] = U_tgt[b * (NN * NN) + tid];

  // Combined edge mask EM (edge must exist in BOTH graphs — the reference's
  // M_e-flattening quirk multiplies src&tgt masks for each edge slot)
  {
    int p = tid >> 4, q = tid & 15;
    float em = 0.0f;
    if (p < q) {
      int e = 15 * p - (p * (p - 1)) / 2 + (q - p - 1);
      em = (A_src[b * EE + e] > 0 && A_tgt[b * EE + e] > 0) ? 1.0f : 0.0f;
    }
    tables[b * 1024 + 768 + tid] = em;
  }

  // Prefetch this wave's weight rows (global_prefetch_b8)
  {
    int arow = wave * 16 + lrow;
    __builtin_prefetch(&l1g[arow * DD], 0, 3);
    __builtin_prefetch(&l2g[arow * DD], 0, 3);
  }
  __syncthreads();

  // ---- Stage 1: wave w -> 16-row tile w of T1 and T2 (K = 128, 32 WMMAs each)
  {
    v8f acc1 = {0.f, 0.f, 0.f, 0.f, 0.f, 0.f, 0.f, 0.f};
    v8f acc2 = {0.f, 0.f, 0.f, 0.f, 0.f, 0.f, 0.f, 0.f};
    const int arow = wave * 16 + lrow;
#pragma unroll 8
    for (int k0 = 0; k0 < DD; k0 += 4) {
      int ka = k0 + 2 * hi;  // A layout: lanes<16 hold K=k0,k0+1; lanes>=16 hold K=k0+2,k0+3
      v2f a1 = {l1g[arow * DD + ka], l1g[arow * DD + ka + 1]};
      v2f a2 = {l2g[arow * DD + ka], l2g[arow * DD + ka + 1]};
      v2f bb = {sFt[ka * NN + lrow], sFt[(ka + 1) * NN + lrow]};
      acc1 = wmma4(a1, bb, acc1);
      acc2 = wmma4(a2, bb, acc2);
    }
    // D layout: VGPR r, lanes<16 -> row r; lanes>=16 -> row r+8
#pragma unroll
    for (int r = 0; r < 8; ++r) {
      sT1[(wave * 16 + r + 8 * hi) * NN + lrow] = acc1[r];
      sT2[(wave * 16 + r + 8 * hi) * NN + lrow] = acc2[r];
    }
  }
  __syncthreads();

  // ---- Stage 2: P1 (wave0), P2 (wave1), Mp (wave2) — wave-uniform branches
  float* Pb = tables + b * 1024;
  if (wave == 0) {
    v8f acc = {0.f, 0.f, 0.f, 0.f, 0.f, 0.f, 0.f, 0.f};
#pragma unroll 8
    for (int k0 = 0; k0 < DD; k0 += 4) {
      int ka = k0 + 2 * hi;
      v2f a  = {sFs[ka * NN + lrow], sFs[(ka + 1) * NN + lrow]};  // (F_src^T)[m][k]
      v2f bb = {sT1[ka * NN + lrow], sT1[(ka + 1) * NN + lrow]};
      acc = wmma4(a, bb, acc);
    }
#pragma unroll
    for (int r = 0; r < 8; ++r) Pb[(r + 8 * hi) * NN + lrow] = acc[r];
  } else if (wave == 1) {
    v8f acc = {0.f, 0.f, 0.f, 0.f, 0.f, 0.f, 0.f, 0.f};
#pragma unroll 8
    for (int k0 = 0; k0 < DD; k0 += 4) {
      int ka = k0 + 2 * hi;
      v2f a  = {sFs[ka * NN + lrow], sFs[(ka + 1) * NN + lrow]};
      v2f bb = {sT2[ka * NN + lrow], sT2[(ka + 1) * NN + lrow]};
      acc = wmma4(a, bb, acc);
    }
#pragma unroll
    for (int r = 0; r < 8; ++r) Pb[256 + (r + 8 * hi) * NN + lrow] = acc[r];
  } else if (wave == 2) {
    v8f acc = {0.f, 0.f, 0.f, 0.f, 0.f, 0.f, 0.f, 0.f};
#pragma unroll
    for (int k0 = 0; k0 < NN; k0 += 4) {
      int ka = k0 + 2 * hi;
      v2f a  = {sUs[ka * NN + lrow], sUs[(ka + 1) * NN + lrow]};  // (U_src^T)[n][d]
      v2f bb = {sUt[ka * NN + lrow], sUt[(ka + 1) * NN + lrow]};
      acc = wmma4(a, bb, acc);
    }
#pragma unroll
    for (int r = 0; r < 8; ++r) Pb[512 + (r + 8 * hi) * NN + lrow] = acc[r];
  }
}

// ---------------------------------------------------------------------------
// Kernel 3: materialize M[16,256,256] with a wide grid (coalesced 4 MiB store).
//   M[b, i2*16+i1, j2*16+j1] = [i==j]*Mp[i2,i1]
//       + EM[i2,j2]*EM[i1,j1]*(P1[i2,i1]+P2[i2,j1]+P2[j2,i1]+P1[j2,j1])
// ---------------------------------------------------------------------------
__global__ void __launch_bounds__(256)
fill_output_kernel(const float* __restrict__ tables, float* __restrict__ out) {
  int gid = blockIdx.x * 256 + threadIdx.x;   // 16*256*256 = 1,048,576
  int b = gid >> 16;
  int i = (gid >> 8) & 255;
  int j = gid & 255;
  const float* Pb = tables + b * 1024;
  int i2 = i >> 4, i1 = i & 15, j2 = j >> 4, j1 = j & 15;
  float val = (i == j) ? Pb[512 + i] : 0.0f;  // diag(Mp_flat)
  if (Pb[768 + i2 * 16 + j2] > 0.5f && Pb[768 + i1 * 16 + j1] > 0.5f) {
    val += Pb[i2 * 16 + i1] + Pb[256 + i2 * 16 + j1] +
           Pb[256 + j2 * 16 + i1] + Pb[j2 * 16 + j1];
  }
  out[gid] = val;
}

// ---------------------------------------------------------------------------
extern "C" void kernel_launch(void* const* d_in, const int* in_sizes, int n_in,
                              void* d_out, int out_size, void* d_ws, size_t ws_size,
                              hipStream_t stream) {
  const int*   A_src = (const int*)d_in[0];
  const int*   A_tgt = (const int*)d_in[1];
  const float* F_src = (const float*)d_in[2];
  const float* F_tgt = (const float*)d_in[3];
  const float* U_src = (const float*)d_in[4];
  const float* U_tgt = (const float*)d_in[5];
  const float* lam1  = (const float*)d_in[6];
  const float* lam2  = (const float*)d_in[7];
  float* out = (float*)d_out;

  // Workspace layout (floats): l1[16384] | l2[16384] | tables[16*1024]
  // => 49152 floats = 192 KiB required of d_ws.
  float* ws     = (float*)d_ws;
  float* l1     = ws;
  float* l2     = ws + 16384;
  float* tables = ws + 32768;

  prep_weights_kernel<<<64, 256, 0, stream>>>(lam1, lam2, l1, l2);
  batch_tables_kernel<<<BATCHN, 256, 0, stream>>>(A_src, A_tgt, F_src, F_tgt,
                                                  U_src, U_tgt, l1, l2, tables);
  fill_output_kernel<<<(BATCHN * 256 * 256) / 256, 256, 0, stream>>>(tables, out);
}